// ChebyshevConvolution_55662776156462
// MI455X (gfx1250) — compile-verified
//
#include <hip/hip_runtime.h>
#include <hip/hip_bf16.h>
#include <stdint.h>

// ChebyshevConvolution forward: N=50000 nodes, E=800000 edges, F=H=64, C=10.
// GEMMs done with V_WMMA_F32_16X16X4_F32 (fp32 matrix core path on gfx1250).
// Weights staged in LDS in WMMA-B pair-interleaved order: one ds_load_b64 per
// B operand, bank-conflict free (row stride mod 64 == 32 splits half-waves
// across bank halves).

#define FDIM 64
#define HDIM 64
#define CDIM 10

#define GWAVES 4                 // waves per GEMM block (128 threads, wave32)
#define GROWS  (GWAVES * 16)     // rows per GEMM block
#define ASTR   68                // LDS A row stride (floats): conflict-free + 16B aligned
#define WPSTR  160               // LDS B k-pair row stride for 64-wide B (128 data + 32 pad)
#define WPROWS 32                // 64 k values -> 32 k-pair rows
#define WOSTR  96                // LDS B k-pair row stride for 16-wide B (32 data + 64 pad)

typedef float v2f __attribute__((ext_vector_type(2)));
typedef float v8f __attribute__((ext_vector_type(8)));

__device__ __forceinline__ v8f wmma4(v2f a, v2f b, v8f c) {
  // D(16x16,f32) = A(16x4,f32) x B(4x16,f32) + C
  // args: (neg_a, A, neg_b, B, c_mod, C, reuse_a, reuse_b)
  return __builtin_amdgcn_wmma_f32_16x16x4_f32(false, a, false, b, (short)0, c,
                                               false, false);
}

// ---------------- graph-prep kernels ----------------

__global__ void zero_kernel(float* __restrict__ p, int n) {
  int i = blockIdx.x * blockDim.x + threadIdx.x;
  if (i < n) p[i] = 0.0f;
}

__global__ void degree_kernel(const int* __restrict__ src, float* __restrict__ deg, int E) {
  int e = blockIdx.x * blockDim.x + threadIdx.x;
  if (e < E) atomicAdd(&deg[src[e]], 1.0f);
}

__global__ void dis_kernel(const float* __restrict__ deg, float* __restrict__ dis, int n) {
  int i = blockIdx.x * blockDim.x + threadIdx.x;
  if (i >= n) return;
  float d = deg[i];
  dis[i] = (d > 0.0f) ? rsqrtf(fmaxf(d, 1.0f)) : 0.0f;
}

__global__ void edgew_kernel(const int* __restrict__ src, const int* __restrict__ dst,
                             const float* __restrict__ dis, float* __restrict__ w, int E) {
  int e = blockIdx.x * blockDim.x + threadIdx.x;
  if (e < E) w[e] = -(dis[src[e]] * dis[dst[e]]);
}

// lap[dst] += w * h[src]; one thread per (edge, float4-chunk of 64 feats)
__global__ void scatter_kernel(const int* __restrict__ src, const int* __restrict__ dst,
                               const float* __restrict__ w, const float* __restrict__ h,
                               float* __restrict__ lap, int E) {
  int t = blockIdx.x * blockDim.x + threadIdx.x;
  int e = t >> 4;
  if (e >= E) return;
  int chunk = t & 15;
  int s = src[e], d = dst[e];
  float we = w[e];
  float4 hv = ((const float4*)(h + (size_t)s * HDIM))[chunk];
  float* lp = lap + (size_t)d * HDIM + chunk * 4;
  atomicAdd(lp + 0, we * hv.x);
  atomicAdd(lp + 1, we * hv.y);
  atomicAdd(lp + 2, we * hv.z);
  atomicAdd(lp + 3, we * hv.w);
}

// ---------------- WMMA GEMM helpers ----------------

// Stage 64x64 W into pair-interleaved B layout:
// sWp[(k>>1)*WPSTR + n*2 + (k&1)] = W[k*64 + n]
__device__ __forceinline__ void stage_w_pairs(float* sWp, const float* __restrict__ W,
                                              int tid) {
  const float4* Wv = (const float4*)W;
#pragma unroll
  for (int i = 0; i < 8; ++i) {
    int idx = tid + i * 128;           // float4 index 0..1023
    float4 v = Wv[idx];
    int k = idx >> 4;                  // row 0..63
    int c4 = (idx & 15) * 4;           // col base
    float* dstp = sWp + (k >> 1) * WPSTR + (k & 1);
    dstp[(c4 + 0) * 2] = v.x;
    dstp[(c4 + 1) * 2] = v.y;
    dstp[(c4 + 2) * 2] = v.z;
    dstp[(c4 + 3) * 2] = v.w;
  }
}

// Stage 16 rows x 64 cols of A into this wave's LDS stripe (coalesced float4).
__device__ __forceinline__ void stage_rows(float* sAw, const float* __restrict__ A,
                                           int rowBase, int n, int lane) {
#pragma unroll
  for (int i = 0; i < 8; ++i) {
    int fi = lane + i * 32;            // 0..255
    int r = fi >> 4, c4 = fi & 15;
    int row = rowBase + r;
    if (row >= n) row = n - 1;         // clamp (WMMA needs full EXEC)
    ((float4*)(sAw + r * ASTR))[c4] = ((const float4*)(A + (size_t)row * 64))[c4];
  }
}

// acc[nt] += A_stripe(16x64) x W(64x64); 16 steps of K=4, one b64 per operand.
__device__ __forceinline__ void mm_accum64(const float* sAw, const float* sWp,
                                           int lane, v8f acc[4]) {
  const int nlo = lane & 15;           // A: M index, B: N index
  const int g = lane >> 4;
#pragma unroll
  for (int kk = 0; kk < 16; ++kk) {
    const int k0 = kk * 4 + 2 * g;
    v2f a = *(const v2f*)(sAw + nlo * ASTR + k0);
    const float* brow = sWp + (kk * 2 + g) * WPSTR;
#pragma unroll
    for (int nt = 0; nt < 4; ++nt) {
      v2f b = *(const v2f*)(brow + (nt * 16 + nlo) * 2);
      acc[nt] = wmma4(a, b, acc[nt]);
    }
  }
}

// ---------------- GEMM kernels ----------------

// out = relu(A[n x 64] @ W[64 x 64] + bias)
__global__ void __launch_bounds__(128)
gemm64_relu_kernel(const float* __restrict__ A, const float* __restrict__ W,
                   const float* __restrict__ bias, float* __restrict__ out, int n) {
  __shared__ float sWp[WPROWS * WPSTR];
  __shared__ float sA[GWAVES * 16 * ASTR];
  __shared__ float sB[64];
  const int tid = threadIdx.x, wave = tid >> 5, lane = tid & 31;
  stage_w_pairs(sWp, W, tid);
  if (tid < 64) sB[tid] = bias[tid];

  const int rowBase = blockIdx.x * GROWS + wave * 16;
  float* sAw = sA + wave * 16 * ASTR;
  stage_rows(sAw, A, rowBase, n, lane);
  __syncthreads();

  v8f z = {0, 0, 0, 0, 0, 0, 0, 0};
  v8f acc[4] = {z, z, z, z};
  mm_accum64(sAw, sWp, lane, acc);

  const int nlo = lane & 15, g = lane >> 4;
#pragma unroll
  for (int nt = 0; nt < 4; ++nt)
#pragma unroll
    for (int r = 0; r < 8; ++r) {
      int row = rowBase + r + 8 * g;
      int col = nt * 16 + nlo;
      if (row < n) out[(size_t)row * 64 + col] = fmaxf(acc[nt][r] + sB[col], 0.0f);
    }
}

// h = relu(h @ W0 + lap @ W1 + bias) + h   (ChebConv K=2 + residual, in place)
__global__ void __launch_bounds__(128)
gemm64_cheb2_kernel(float* __restrict__ h, const float* __restrict__ lap,
                    const float* __restrict__ W0, const float* __restrict__ W1,
                    const float* __restrict__ bias, int n) {
  __shared__ float sWp[2 * WPROWS * WPSTR];
  __shared__ float sA[GWAVES * 16 * ASTR];
  __shared__ float sB[64];
  const int tid = threadIdx.x, wave = tid >> 5, lane = tid & 31;
  stage_w_pairs(sWp, W0, tid);
  stage_w_pairs(sWp + WPROWS * WPSTR, W1, tid);
  if (tid < 64) sB[tid] = bias[tid];

  const int rowBase = blockIdx.x * GROWS + wave * 16;
  float* sAw = sA + wave * 16 * ASTR;

  v8f z = {0, 0, 0, 0, 0, 0, 0, 0};
  v8f acc[4] = {z, z, z, z};

  stage_rows(sAw, h, rowBase, n, lane);             // T0 = h
  __syncthreads();
  mm_accum64(sAw, sWp, lane, acc);                  // += h @ W0
  __syncthreads();
  stage_rows(sAw, lap, rowBase, n, lane);           // T1 = L_hat h
  __syncthreads();
  mm_accum64(sAw, sWp + WPROWS * WPSTR, lane, acc); // += lap @ W1

  const int nlo = lane & 15, g = lane >> 4;
#pragma unroll
  for (int nt = 0; nt < 4; ++nt)
#pragma unroll
    for (int r = 0; r < 8; ++r) {
      int row = rowBase + r + 8 * g;
      int col = nt * 16 + nlo;
      if (row < n) {
        size_t idx = (size_t)row * 64 + col;
        float v = fmaxf(acc[nt][r] + sB[col], 0.0f);
        h[idx] = v + h[idx];                        // residual, in place (row-owned)
      }
    }
}

// logits[n x 16] = A[n x 64] @ Wl(64x10 zero-padded to 64x16) + bl
__global__ void __launch_bounds__(128)
gemm_out_kernel(const float* __restrict__ A, const float* __restrict__ Wl,
                const float* __restrict__ bl, float* __restrict__ logits, int n) {
  __shared__ float sWo[WPROWS * WOSTR];
  __shared__ float sA[GWAVES * 16 * ASTR];
  __shared__ float sB[16];
  const int tid = threadIdx.x, wave = tid >> 5, lane = tid & 31;
  for (int idx = tid; idx < 64 * 16; idx += 128) {
    int k = idx >> 4, c = idx & 15;
    float v = (c < CDIM) ? Wl[k * CDIM + c] : 0.0f;
    sWo[(k >> 1) * WOSTR + c * 2 + (k & 1)] = v;
  }
  if (tid < 16) sB[tid] = (tid < CDIM) ? bl[tid] : 0.0f;

  const int rowBase = blockIdx.x * GROWS + wave * 16;
  float* sAw = sA + wave * 16 * ASTR;
  stage_rows(sAw, A, rowBase, n, lane);
  __syncthreads();

  const int nlo = lane & 15, g = lane >> 4;
  v8f acc = {0, 0, 0, 0, 0, 0, 0, 0};
#pragma unroll
  for (int kk = 0; kk < 16; ++kk) {
    const int k0 = kk * 4 + 2 * g;
    v2f a = *(const v2f*)(sAw + nlo * ASTR + k0);
    v2f b = *(const v2f*)(sWo + (kk * 2 + g) * WOSTR + nlo * 2);
    acc = wmma4(a, b, acc);
  }
#pragma unroll
  for (int r = 0; r < 8; ++r) {
    int row = rowBase + r + 8 * g;
    if (row < n) logits[(size_t)row * 16 + nlo] = acc[r] + sB[nlo];
  }
}

// row-wise log_softmax over first 10 of 16 padded logits
__global__ void logsoftmax_kernel(const float* __restrict__ logits,
                                  float* __restrict__ out, int n) {
  int i = blockIdx.x * blockDim.x + threadIdx.x;
  if (i >= n) return;
  const float* v = logits + (size_t)i * 16;
  float m = v[0];
#pragma unroll
  for (int c = 1; c < CDIM; ++c) m = fmaxf(m, v[c]);
  float s = 0.0f;
#pragma unroll
  for (int c = 0; c < CDIM; ++c) s += expf(v[c] - m);
  float ls = logf(s);
#pragma unroll
  for (int c = 0; c < CDIM; ++c) out[(size_t)i * CDIM + c] = v[c] - m - ls;
}

// ---------------- host launch ----------------

extern "C" void kernel_launch(void* const* d_in, const int* in_sizes, int n_in,
                              void* d_out, int out_size, void* d_ws, size_t ws_size,
                              hipStream_t stream) {
  const float* x    = (const float*)d_in[0];
  const int*   ei   = (const int*)d_in[1];   // [2, E] (integer -> int32 per harness)
  const float* W1_0 = (const float*)d_in[2];
  const float* b1   = (const float*)d_in[3];
  const float* W2_0 = (const float*)d_in[4];
  const float* W2_1 = (const float*)d_in[5];
  const float* b2   = (const float*)d_in[6];
  const float* Wl   = (const float*)d_in[7];
  const float* bl   = (const float*)d_in[8];

  const int N = in_sizes[0] / FDIM;   // 50000
  const int E = in_sizes[1] / 2;      // 800000
  const int* src = ei;
  const int* dst = ei + E;

  // workspace layout (floats): deg[N] | lap[N*64] | dis[N] | w[E] | h[N*64] | logits[N*16]
  float* ws     = (float*)d_ws;
  float* deg    = ws;
  float* lap    = deg + N;                 // contiguous with deg -> zero both at once
  float* dis    = lap + (size_t)N * 64;
  float* wbuf   = dis + N;
  float* h      = wbuf + E;
  float* logits = h + (size_t)N * 64;
  float* out    = (float*)d_out;

  // 1) zero deg + lap
  {
    int tot = N * 65;
    zero_kernel<<<(tot + 255) / 256, 256, 0, stream>>>(deg, tot);
  }
  // 2) degree / dis / edge weights
  degree_kernel<<<(E + 255) / 256, 256, 0, stream>>>(src, deg, E);
  dis_kernel<<<(N + 255) / 256, 256, 0, stream>>>(deg, dis, N);
  edgew_kernel<<<(E + 255) / 256, 256, 0, stream>>>(src, dst, dis, wbuf, E);

  const int gblocks = (N + GROWS - 1) / GROWS;
  // 3) h = relu(x @ W1_0 + b1)
  gemm64_relu_kernel<<<gblocks, 128, 0, stream>>>(x, W1_0, b1, h, N);
  // 4) lap = segment_sum(w * h[src] -> dst)
  {
    long long tot = (long long)E * 16;
    scatter_kernel<<<(int)((tot + 255) / 256), 256, 0, stream>>>(src, dst, wbuf, h, lap, E);
  }
  // 5) h = relu(h@W2_0 + lap@W2_1 + b2) + h
  gemm64_cheb2_kernel<<<gblocks, 128, 0, stream>>>(h, lap, W2_0, W2_1, b2, N);
  // 6) logits = h @ Wl + bl (padded to 16 cols)
  gemm_out_kernel<<<gblocks, 128, 0, stream>>>(h, Wl, bl, logits, N);
  // 7) log_softmax -> d_out [N x 10]
  logsoftmax_kernel<<<(N + 255) / 256, 256, 0, stream>>>(logits, out, N);
}